// MoELayerTaskAware_33320356282923
// MI455X (gfx1250) — compile-verified
//
#include <hip/hip_runtime.h>
#include <stdint.h>

typedef float v2f __attribute__((ext_vector_type(2)));
typedef float v8f __attribute__((ext_vector_type(8)));

#define NUM_EXPERTS 8
#define NUM_TASKS   8
#define D_IN        1024
#define D_HID       4096
#define D_OUT       1024
#define NTOK        8192

#define MTILE 32      // tokens per workgroup
#define JC    64      // hidden-dim chunk (K of GEMM2, N of GEMM1)
#define KCC   64      // k chunk for GEMM1
#define OGR   128     // output rows staged per w2 LDS tile

// LDS strides in dwords, padded (+4) => bank = 4*(lane&15), conflict-free b64 reads.
// Row stride in bytes = 272 = 16*17, so every float4 slot stays 16B-aligned for b128.
#define XP_S 68
#define W1_S 68
#define H_S  68
#define W2_S 68

// ---- CDNA5 async global->LDS copy (ASYNCcnt-tracked, bypasses VGPR staging) ----
__device__ __forceinline__ void async_ld16(uint32_t lds_off, const float* base, uint32_t goff) {
  // GVS mode: 64-bit SGPR base + 32-bit per-lane byte offset; vdst = LDS byte offset
  asm volatile("global_load_async_to_lds_b128 %0, %1, %2"
               :: "v"(lds_off), "v"(goff), "s"(base)
               : "memory");
}
__device__ __forceinline__ void wait_async0() {
  asm volatile("s_wait_asynccnt 0" ::: "memory");
}
#define LDSOFF(p) ((uint32_t)(uintptr_t)(p))

__global__ __launch_bounds__(256) void moe_fused(
    const float* __restrict__ x,    // [NTOK][D_IN]
    const float* __restrict__ gw,   // [T][E][D_IN]
    const float* __restrict__ w1,   // [E][D_HID][D_IN]
    const float* __restrict__ b1,   // [E][D_HID]
    const float* __restrict__ w2,   // [E][D_OUT][D_HID]
    const float* __restrict__ b2,   // [E][D_OUT]
    float* __restrict__ out)        // [NTOK][D_OUT]
{
  __shared__ float xp [2][MTILE * XP_S];      // 2 x  8704 B  (GEMM1 A panels)
  __shared__ float w1s[2][JC * W1_S];         // 2 x 17408 B  (GEMM1 B tiles)
  __shared__ float hs [MTILE * H_S];          //      8704 B  (scaled relu(h))
  __shared__ float w2s[2][OGR * W2_S];        // 2 x 34816 B  (GEMM2 B tiles)
  __shared__ float cs [MTILE * NUM_EXPERTS];  //      1024 B  (combine weights)
  // total: 131584 B

  const int tid  = threadIdx.x;
  const int wave = tid >> 5;
  const int lane = tid & 31;
  const int l15  = lane & 15;
  const int hi   = lane >> 4;          // 0/1 half-wave
  const int n0   = blockIdx.x * MTILE;

  // ---------------- gating: combine weights for our 32 tokens ----------------
  for (int mm = 0; mm < 4; ++mm) {
    const int m = wave * 4 + mm;
    const int n = n0 + m;
    const int t = n & (NUM_TASKS - 1);
    const float* xr = x + (size_t)n * D_IN;
    float logit[NUM_EXPERTS];
#pragma unroll
    for (int e = 0; e < NUM_EXPERTS; ++e) {
      const float* gr = gw + ((size_t)t * NUM_EXPERTS + e) * D_IN;
      float s = 0.f;
      for (int i = 0; i < D_IN / 32; ++i) {
        const int d = lane + i * 32;
        s += xr[d] * gr[d];
      }
#pragma unroll
      for (int off = 16; off > 0; off >>= 1) s += __shfl_xor(s, off);
      logit[e] = s;
    }
    int i0 = 0; float v0 = logit[0];
#pragma unroll
    for (int e = 1; e < NUM_EXPERTS; ++e)
      if (logit[e] > v0) { v0 = logit[e]; i0 = e; }
    int i1 = -1; float v1 = -3.4e38f;
#pragma unroll
    for (int e = 0; e < NUM_EXPERTS; ++e)
      if (e != i0 && logit[e] > v1) { v1 = logit[e]; i1 = e; }
    const float wa = 1.f / (1.f + __expf(v1 - v0));
    const float wb = 1.f - wa;
    if (lane < NUM_EXPERTS)
      cs[m * NUM_EXPERTS + lane] = (lane == i0) ? wa : ((lane == i1) ? wb : 0.f);
  }
  __syncthreads();

  // per-thread issue slots for the async staging
  const int xrow = tid >> 4;           // used with i*16 rows per pass
  const int c4   = tid & 15;           // float4 column within a 64-float row

  // ---------------- fused dense MoE, output tile resident in VGPRs ----------
  v8f acc[16];
#pragma unroll
  for (int i = 0; i < 16; ++i) acc[i] = (v8f){0,0,0,0,0,0,0,0};

  const int mtA = wave & 1;            // GEMM1 wave -> (m-tile, j-tile)
  const int jt  = wave >> 1;

  for (int e = 0; e < NUM_EXPERTS; ++e) {
    const float* w1e = w1 + (size_t)e * D_HID * D_IN;
    const float* w2e = w2 + (size_t)e * D_OUT * D_HID;

    float cw1[8];
#pragma unroll
    for (int r = 0; r < 8; ++r)
      cw1[r] = cs[(mtA * 16 + r + 8 * hi) * NUM_EXPERTS + e];

    for (int jc = 0; jc < D_HID; jc += JC) {
      // ====== GEMM1: h[32][64] = x[32][1024] * w1[jc:jc+64][:]^T ======
      v8f hacc = (v8f){0,0,0,0,0,0,0,0};

      // issue k-panel 0 into buffer 0 (async, straight to LDS)
      {
#pragma unroll
        for (int i = 0; i < 2; ++i) {
          const int row = xrow + i * 16;                     // 0..31
          async_ld16(LDSOFF(&xp[0][0]) + (uint32_t)(row * XP_S + c4 * 4) * 4u,
                     x, (uint32_t)(((n0 + row) * D_IN + 0 + c4 * 4) * 4));
        }
#pragma unroll
        for (int i = 0; i < 4; ++i) {
          const int row = xrow + i * 16;                     // 0..63
          async_ld16(LDSOFF(&w1s[0][0]) + (uint32_t)(row * W1_S + c4 * 4) * 4u,
                     w1e, (uint32_t)(((jc + row) * D_IN + 0 + c4 * 4) * 4));
        }
      }

      for (int kc16 = 0; kc16 < D_IN / KCC; ++kc16) {
        wait_async0();
        __syncthreads();
        const int nkc = kc16 + 1;
        if (nkc < D_IN / KCC) {
          const int kn = nkc * KCC;
          const int nb = nkc & 1;
#pragma unroll
          for (int i = 0; i < 2; ++i) {
            const int row = xrow + i * 16;
            async_ld16(LDSOFF(&xp[nb][0]) + (uint32_t)(row * XP_S + c4 * 4) * 4u,
                       x, (uint32_t)(((n0 + row) * D_IN + kn + c4 * 4) * 4));
          }
#pragma unroll
          for (int i = 0; i < 4; ++i) {
            const int row = xrow + i * 16;
            async_ld16(LDSOFF(&w1s[nb][0]) + (uint32_t)(row * W1_S + c4 * 4) * 4u,
                       w1e, (uint32_t)(((jc + row) * D_IN + kn + c4 * 4) * 4));
          }
        }
        const float* xb  = &xp [kc16 & 1][0];
        const float* w1b = &w1s[kc16 & 1][0];
#pragma unroll
        for (int ks = 0; ks < KCC; ks += 4) {
          const v2f a = *(const v2f*)&xb [(mtA * 16 + l15) * XP_S + ks + 2 * hi];
          const v2f b = *(const v2f*)&w1b[(jt  * 16 + l15) * W1_S + ks + 2 * hi];
          hacc = __builtin_amdgcn_wmma_f32_16x16x4_f32(
                     false, a, false, b, (short)0, hacc, false, false);
        }
      }

      // prefetch GEMM2 tile og=0 into buffer 0 while we do the h epilogue
      {
#pragma unroll
        for (int i = 0; i < 8; ++i) {
          const int row = xrow + i * 16;                     // 0..127
          async_ld16(LDSOFF(&w2s[0][0]) + (uint32_t)(row * W2_S + c4 * 4) * 4u,
                     w2e, (uint32_t)((row * D_HID + jc + c4 * 4) * 4));
        }
      }

      // bias + relu + combine-scale, write h tile to LDS
      const float b1v = b1[e * D_HID + jc + jt * 16 + l15];
      __syncthreads();  // previous jc's GEMM2 readers of hs are done
#pragma unroll
      for (int r = 0; r < 8; ++r) {
        float v = hacc[r] + b1v;
        v = v > 0.f ? v : 0.f;
        hs[(mtA * 16 + r + 8 * hi) * H_S + jt * 16 + l15] = v * cw1[r];
      }
      __syncthreads();

      // ====== GEMM2: out[32][1024] += h[32][64] * w2[:, jc:jc+64]^T ======
#pragma unroll
      for (int og = 0; og < D_OUT / OGR; ++og) {             // 8 groups of 128
        wait_async0();
        __syncthreads();
        if (og + 1 < D_OUT / OGR) {
          const int nb = (og + 1) & 1;
          const int ob = (og + 1) * OGR;
#pragma unroll
          for (int i = 0; i < 8; ++i) {
            const int row = xrow + i * 16;
            async_ld16(LDSOFF(&w2s[nb][0]) + (uint32_t)(row * W2_S + c4 * 4) * 4u,
                       w2e, (uint32_t)(((ob + row) * D_HID + jc + c4 * 4) * 4));
          }
        }
        const float* w2b = &w2s[og & 1][0];
#pragma unroll
        for (int mt2 = 0; mt2 < 2; ++mt2) {
          v8f c = acc[og * 2 + mt2];
#pragma unroll
          for (int js = 0; js < JC; js += 4) {
            const v2f a = *(const v2f*)&hs [(mt2  * 16 + l15) * H_S  + js + 2 * hi];
            const v2f b = *(const v2f*)&w2b[(wave * 16 + l15) * W2_S + js + 2 * hi];
            c = __builtin_amdgcn_wmma_f32_16x16x4_f32(
                    false, a, false, b, (short)0, c, false, false);
          }
          acc[og * 2 + mt2] = c;
        }
      }
    }
  }

  // ---------------- epilogue: + sum_e c[m][e]*b2[e][col], store ----------------
#pragma unroll
  for (int og = 0; og < D_OUT / OGR; ++og) {
    const int col = og * OGR + wave * 16 + l15;
    float b2v[NUM_EXPERTS];
#pragma unroll
    for (int e = 0; e < NUM_EXPERTS; ++e) b2v[e] = b2[e * D_OUT + col];
#pragma unroll
    for (int mt2 = 0; mt2 < 2; ++mt2) {
      const v8f c = acc[og * 2 + mt2];
#pragma unroll
      for (int r = 0; r < 8; ++r) {
        const int m = mt2 * 16 + r + 8 * hi;
        float s = 0.f;
#pragma unroll
        for (int e = 0; e < NUM_EXPERTS; ++e)
          s += cs[m * NUM_EXPERTS + e] * b2v[e];
        out[(size_t)(n0 + m) * D_OUT + col] = c[r] + s;
      }
    }
  }
}

extern "C" void kernel_launch(void* const* d_in, const int* in_sizes, int n_in,
                              void* d_out, int out_size, void* d_ws, size_t ws_size,
                              hipStream_t stream) {
  (void)in_sizes; (void)n_in; (void)out_size; (void)d_ws; (void)ws_size;
  const float* x  = (const float*)d_in[0];
  const float* gw = (const float*)d_in[1];
  const float* w1 = (const float*)d_in[2];
  const float* b1 = (const float*)d_in[3];
  const float* w2 = (const float*)d_in[4];
  const float* b2 = (const float*)d_in[5];
  float* o = (float*)d_out;
  dim3 grid(NTOK / MTILE);   // 256 workgroups
  dim3 block(256);           // 8 waves (wave32)
  hipLaunchKernelGGL(moe_fused, grid, block, 0, stream, x, gw, w1, b1, w2, b2, o);
}